// TreeNode_57518202028582
// MI455X (gfx1250) — compile-verified
//
#include <hip/hip_runtime.h>
#include <math.h>

// Soft decision tree, fused on gfx1250:
//   phase 1: fp32 GEMM via V_WMMA_F32_16X16X4_F32 -> scores kept in LDS heap
//            (4 independent WMMA accumulator chains per wave for XDL ILP)
//   phase 2: bottom-up tree expectation in LDS -> 1 scalar per row
//
// Shapes: x (16384,1024) f32; W_split (1023,1024); b_split (1023);
//         W_leaf (1024,1024); b_leaf (1024); out (16384,1) f32.

typedef __attribute__((ext_vector_type(2))) float v2f;
typedef __attribute__((ext_vector_type(4))) float v4f;
typedef __attribute__((ext_vector_type(8))) float v8f;

#define INPUT_DIM 1024
#define BATCH     16384
#define N_SLOTS   2048          // heap: slot 0 unused, 1..1023 internal, 1024..2047 leaves
#define M_TILE    16            // batch rows per workgroup
#define XS_STRIDE 1028          // 1024 + 4 pad -> bank-conflict-free A-fragment loads
#define S_STRIDE  2048
#define THREADS   256
#define SMEM_BYTES ((M_TILE * XS_STRIDE + M_TILE * S_STRIDE) * sizeof(float))  // 64K + 128K

__device__ __forceinline__ float fast_sigmoid(float v) {
    // v_exp_f32 + v_add + v_rcp_f32 (vs ~12-op IEEE division sequence)
    return __builtin_amdgcn_rcpf(1.0f + __expf(-v));
}

__launch_bounds__(THREADS, 1)
__global__ void tree_fused_wmma_f32(const float* __restrict__ x,
                                    const float* __restrict__ Wsplit,
                                    const float* __restrict__ bsplit,
                                    const float* __restrict__ Wleaf,
                                    const float* __restrict__ bleaf,
                                    float* __restrict__ out)
{
    extern __shared__ float smem[];
    float* xs = smem;                           // [M_TILE][XS_STRIDE]
    float* S  = smem + M_TILE * XS_STRIDE;      // [M_TILE][S_STRIDE] score heap

    const int tid  = threadIdx.x;
    const int lane = tid & 31;
    const int wave = tid >> 5;                  // 8 waves
    const int half = lane >> 4;                 // 0: K={0,1}, 1: K={2,3}
    const int lsub = lane & 15;                 // M (for A) / N (for B) index
    const int m0   = blockIdx.x * M_TILE;

    // ---- stage x tile: 16 rows x 1024 f32 (64 KB); NT loads (x is read once
    //      per grid) so the L2 stays dedicated to the W matrices ----
    for (int i = tid; i < M_TILE * (INPUT_DIM / 4); i += THREADS) {
        const int r  = i >> 8;                  // 256 float4 per row
        const int c4 = i & 255;
        *(v4f*)(xs + r * XS_STRIDE + c4 * 4) =
            __builtin_nontemporal_load((const v4f*)(x + (size_t)(m0 + r) * INPUT_DIM + c4 * 4));
    }
    __syncthreads();

    // per-lane A-fragment base: x[m0 + lsub][2*half + k]
    const float* aptr = xs + lsub * XS_STRIDE + 2 * half;

    // ---- GEMM: 64 chunk-pairs of 32 output slots, round-robin over 8 waves.
    //      Per wave: 2 N-chunks share the A fragment; each N-chunk keeps 2
    //      K-split accumulators -> 4 independent v_wmma chains in flight. ----
    for (int cp = wave; cp < N_SLOTS / 32; cp += 8) {
        const int s0 = cp * 32 + lsub;          // heap slot, chunk 0
        const int s1 = s0 + 16;                 // heap slot, chunk 1

        const float *wrow0, *wrow1;
        float bias0, bias1;
        if (s0 == 0)        { wrow0 = Wsplit;                                bias0 = 0.0f; }
        else if (s0 < 1024) { wrow0 = Wsplit + (size_t)(s0 - 1)    * INPUT_DIM; bias0 = bsplit[s0 - 1];    }
        else                { wrow0 = Wleaf  + (size_t)(s0 - 1024) * INPUT_DIM; bias0 = bleaf[s0 - 1024];  }
        if (s1 < 1024)      { wrow1 = Wsplit + (size_t)(s1 - 1)    * INPUT_DIM; bias1 = bsplit[s1 - 1];    }
        else                { wrow1 = Wleaf  + (size_t)(s1 - 1024) * INPUT_DIM; bias1 = bleaf[s1 - 1024];  }
        const float* bptr0 = wrow0 + 2 * half;
        const float* bptr1 = wrow1 + 2 * half;

        v8f accA0 = {}, accA1 = {};             // chunk 0, K even/odd steps
        v8f accB0 = {}, accB1 = {};             // chunk 1, K even/odd steps
        #pragma unroll 4
        for (int k = 0; k < INPUT_DIM; k += 8) {
            v2f a0  = *(const v2f*)(aptr  + k);       // ds_load_2addr_b64 pair
            v2f a1  = *(const v2f*)(aptr  + k + 4);
            v2f b00 = *(const v2f*)(bptr0 + k);       // global_load_b64 x4
            v2f b01 = *(const v2f*)(bptr0 + k + 4);
            v2f b10 = *(const v2f*)(bptr1 + k);
            v2f b11 = *(const v2f*)(bptr1 + k + 4);
            accA0 = __builtin_amdgcn_wmma_f32_16x16x4_f32(false, a0, false, b00, (short)0, accA0, false, false);
            accB0 = __builtin_amdgcn_wmma_f32_16x16x4_f32(false, a0, false, b10, (short)0, accB0, false, false);
            accA1 = __builtin_amdgcn_wmma_f32_16x16x4_f32(false, a1, false, b01, (short)0, accA1, false, false);
            accB1 = __builtin_amdgcn_wmma_f32_16x16x4_f32(false, a1, false, b11, (short)0, accB1, false, false);
        }
        v8f accA = accA0 + accA1;
        v8f accB = accB0 + accB1;

        // writeback C tiles -> LDS heap; VGPR j holds (M = j + 8*half, N = lsub)
        const bool split0 = (s0 >= 1) && (s0 < 1024);
        const bool split1 = (s1 < 1024);
        #pragma unroll
        for (int j = 0; j < 8; ++j) {
            const int row = j + 8 * half;
            float vA = accA[j] + bias0;
            float vB = accB[j] + bias1;
            if (split0) vA = fast_sigmoid(vA);
            if (split1) vB = fast_sigmoid(vB);
            S[row * S_STRIDE + s0] = vA;
            S[row * S_STRIDE + s1] = vB;
        }
    }
    __syncthreads();

    // ---- bottom-up tree expectation: S[s] = (1-p)*S[2s] + p*S[2s+1] ----
    for (int L = 9; L >= 0; --L) {
        const int cnt = M_TILE << L;
        for (int i = tid; i < cnt; i += THREADS) {
            const int row  = i >> L;
            const int slot = (1 << L) + (i & ((1 << L) - 1));
            float* Sr = S + row * S_STRIDE;
            const float p  = Sr[slot];
            const float lv = Sr[2 * slot];
            const float rv = Sr[2 * slot + 1];
            Sr[slot] = fmaf(p, rv - lv, lv);
        }
        __syncthreads();
    }

    if (tid < M_TILE) out[m0 + tid] = S[tid * S_STRIDE + 1];
}

extern "C" void kernel_launch(void* const* d_in, const int* in_sizes, int n_in,
                              void* d_out, int out_size, void* d_ws, size_t ws_size,
                              hipStream_t stream) {
    (void)in_sizes; (void)n_in; (void)d_ws; (void)ws_size; (void)out_size;
    const float* x      = (const float*)d_in[0];
    const float* Wsplit = (const float*)d_in[1];
    const float* bsplit = (const float*)d_in[2];
    const float* Wleaf  = (const float*)d_in[3];
    const float* bleaf  = (const float*)d_in[4];
    float* out = (float*)d_out;

    dim3 grid(BATCH / M_TILE);   // 1024 workgroups
    dim3 block(THREADS);
    tree_fused_wmma_f32<<<grid, block, SMEM_BYTES, stream>>>(
        x, Wsplit, bsplit, Wleaf, bleaf, out);
}